// PLIFNode_39273180954942
// MI455X (gfx1250) — compile-verified
//
#include <hip/hip_runtime.h>

// PLIF spiking-neuron scan over T, elementwise in (B,C,H,W).
// x: [B=32, T=16, C=128, H=32, W=32] fp32 ; w: scalar ; out: spikes, same shape as x.
//
// Pure HBM-streaming kernel: 512 MiB total traffic -> ~23 us floor @ 23.3 TB/s.
// Strategy: 1 thread per 4 consecutive elements (b128 vector memory ops),
// all 16 time-step loads issued before the recurrence (independent of the
// mem chain -> 16 outstanding loads/thread), non-temporal hints since the
// stream (512 MiB) exceeds L2 (192 MB) and every byte is touched once.

typedef float v4f __attribute__((ext_vector_type(4)));

constexpr int   T_STEPS = 16;
constexpr int   CHW     = 128 * 32 * 32;   // 131072 = 2^17
constexpr float VTH     = 0.5f;

__global__ __launch_bounds__(256) void plif_scan_kernel(
    const float* __restrict__ x,
    const float* __restrict__ wparam,
    float* __restrict__ out)
{
    // tau = sigmoid(w); w is a uniform scalar -> scalarized load.
    const float tau = 1.0f / (1.0f + expf(-wparam[0]));

    const int tid  = blockIdx.x * blockDim.x + threadIdx.x;  // [0, B*CHW/4)
    const int s    = tid << 2;            // flat spatial element index (mult. of 4)
    const int b    = s >> 17;             // s / CHW
    const int r    = s & (CHW - 1);       // s % CHW
    const int base = (b << 21) + r;       // b*T*CHW + r   (T*CHW = 2^21)
    // max base + 15*CHW = 67,108,860 < 2^31 -> 32-bit indexing is safe.

    // Stage all T inputs: 16 independent global_load_b128, non-temporal.
    v4f xs[T_STEPS];
#pragma unroll
    for (int t = 0; t < T_STEPS; ++t) {
        xs[t] = __builtin_nontemporal_load(
            reinterpret_cast<const v4f*>(x + base + t * CHW));
    }

    // Sequential PLIF recurrence held entirely in VGPRs.
    v4f mem = {0.0f, 0.0f, 0.0f, 0.0f};
#pragma unroll
    for (int t = 0; t < T_STEPS; ++t) {
        v4f spike;
#pragma unroll
        for (int k = 0; k < 4; ++k) {
            const float m    = tau * mem[k] + xs[t][k];  // mem = tau*mem + x_t
            const bool  fire = (m > VTH);                // spike = H(mem - VTH)
            spike[k]         = fire ? 1.0f : 0.0f;
            mem[k]           = fire ? 0.0f : m;          // hard reset
        }
        __builtin_nontemporal_store(
            spike, reinterpret_cast<v4f*>(out + base + t * CHW));
    }
}

extern "C" void kernel_launch(void* const* d_in, const int* in_sizes, int n_in,
                              void* d_out, int out_size, void* d_ws, size_t ws_size,
                              hipStream_t stream)
{
    const float* x = (const float*)d_in[0];   // [B,T,C,H,W] fp32
    const float* w = (const float*)d_in[1];   // scalar fp32
    float*       o = (float*)d_out;           // [B,T,C,H,W] fp32 spikes

    // threads = B*C*H*W / 4 ; out_size = B*T*C*H*W
    const int total_threads = out_size / T_STEPS / 4;   // 1,048,576
    const int block = 256;
    const int grid  = total_threads / block;            // 4096

    plif_scan_kernel<<<grid, block, 0, stream>>>(x, w, o);
}